// SecondOrderFeatureInteraction_755914244422
// MI455X (gfx1250) — compile-verified
//
#include <hip/hip_runtime.h>

// Gram-matrix upper-triangle (DLRM 2nd-order feature interaction) for MI455X.
// One wave32 per batch element; f32 WMMA 16x16x4; memory-bound (~21us floor).

typedef __attribute__((ext_vector_type(2))) float v2f;
typedef __attribute__((ext_vector_type(8))) float v8f;

#define NFEA 27
#define DDIM 128
#define NPAIR ((NFEA * (NFEA - 1)) / 2)  // 351

__device__ __forceinline__ int pair_idx(int i, int j) {
  // strict upper triangle (j > i), row-major packing
  return i * (NFEA - 1) - (i * (i - 1)) / 2 + (j - i - 1);
}

__global__ __launch_bounds__(256) void soi_gram_wmma_kernel(
    const float* __restrict__ X, float* __restrict__ out, int nbatch) {
  const int lane = threadIdx.x & 31;
  const int wave = threadIdx.x >> 5;
  const int b = blockIdx.x * 8 + wave;
  if (b >= nbatch) return;  // wave-uniform branch: EXEC stays all-1s for live waves

  const float* Xb = X + (size_t)b * (NFEA * DDIM);

  // f32 16x16x4 operand layout: lane l, vgpr v holds elem [l%16][(l>>4)*2 + v].
  // For Gram (B = X^T) the A-layout and B-layout VGPR images coincide, so one
  // loaded fragment feeds both WMMA source operands.
  const int half = lane >> 4;  // selects K pair {0,1} vs {2,3}
  const int r = lane & 15;     // M (and N) index within a 16-wide tile

  const float* row0 = Xb + r * DDIM + half * 2;                 // rows 0..15
  const bool valid1 = (16 + r) < NFEA;                          // rows 16..26
  const float* row1 = Xb + (valid1 ? (16 + r) : 0) * DDIM + half * 2;
  const float s1 = valid1 ? 1.0f : 0.0f;                        // zero pad rows 27..31

  v8f acc00 = {};  // tile (0,0): rows 0-15  x rows 0-15
  v8f acc01 = {};  // tile (0,1): rows 0-15  x rows 16-31
  v8f acc11 = {};  // tile (1,1): rows 16-31 x rows 16-31

#pragma unroll 8
  for (int k = 0; k < DDIM; k += 4) {
    v2f a0 = *(const v2f*)(row0 + k);
    v2f a1 = *(const v2f*)(row1 + k);
    a1 *= s1;  // zero out padding rows before they mix across lanes

    acc00 = __builtin_amdgcn_wmma_f32_16x16x4_f32(
        false, a0, false, a0, (short)0, acc00, false, false);
    acc01 = __builtin_amdgcn_wmma_f32_16x16x4_f32(
        false, a0, false, a1, (short)0, acc01, false, false);
    acc11 = __builtin_amdgcn_wmma_f32_16x16x4_f32(
        false, a1, false, a1, (short)0, acc11, false, false);
  }

  // C/D layout: vgpr v, lanes 0-15 -> M=v, lanes 16-31 -> M=v+8; N = lane%16.
  float* outb = out + (size_t)b * NPAIR;
#pragma unroll
  for (int v = 0; v < 8; ++v) {
    const int m = half * 8 + v;
    // tile (0,0): (i, j) = (m, r), keep strict upper
    if (r > m) outb[pair_idx(m, r)] = acc00[v];
    // tile (0,1): (i, j) = (m, 16 + r); j > i always; need j < 27
    if (r < NFEA - 16) outb[pair_idx(m, 16 + r)] = acc01[v];
    // tile (1,1): (i, j) = (16 + m, 16 + r); need j < 27 and j > i
    if (r < NFEA - 16 && r > m) outb[pair_idx(16 + m, 16 + r)] = acc11[v];
  }
}

extern "C" void kernel_launch(void* const* d_in, const int* in_sizes, int n_in,
                              void* d_out, int out_size, void* d_ws, size_t ws_size,
                              hipStream_t stream) {
  const float* X = (const float*)d_in[0];
  float* out = (float*)d_out;
  const int nbatch = in_sizes[0] / (NFEA * DDIM);  // 32768
  const int nblocks = (nbatch + 7) / 8;            // 8 waves (batches) per block
  soi_gram_wmma_kernel<<<nblocks, 256, 0, stream>>>(X, out, nbatch);
}